// GlobalShift2dV2Portion_16930761081418
// MI455X (gfx1250) — compile-verified
//
#include <hip/hip_runtime.h>

// GlobalShift2dV2, x:(B=16, C=512, H=64, W=64) fp32, SCALE=(4,4), PORTION=0.5
//
// Derivation: channels [0,256) are copied verbatim. For channels [256,512),
// the 64x64 image is a 4x4 grid of 16x16 tiles, tile index k = i0*4 + j0
// (h = i0*16 + i1, w = j0*16 + j1). With channel group g = (c-256)>>4:
//     out[b, c, tile k] = in[b, c, tile (g + k) & 15]   (same i1, j1)
// Pure permutation of whole 16x16 tiles -> both source and destination are
// contiguous at 64-byte (16-float) granularity. One 64B segment per thread:
// 4x global_load_b128 + 4x nontemporal global_store_b128 (th:TH_STORE_NT),
// keeping the 192MB L2 free for the 128MB input across graph replays.

typedef __attribute__((ext_vector_type(4))) float v4f;

__global__ __launch_bounds__(256) void global_shift2d_v2_kernel(
    const float* __restrict__ in, float* __restrict__ out) {
  // One thread = one 16-float w-segment. Total segments = 16*512*64*4 = 2^21.
  const uint32_t seg = blockIdx.x * 256u + threadIdx.x;

  const uint32_t j0p = seg & 3u;           // output w-tile index (0..3)
  const uint32_t h   = (seg >> 2) & 63u;   // output row
  const uint32_t c   = (seg >> 8) & 511u;  // channel
  const uint32_t bc  = seg >> 8;           // b*512 + c (flat)
  const uint32_t outOff = seg << 4;        // float offset of this segment

  uint32_t inOff = outOff;                 // identity for c < 256
  if (c >= 256u) {
    const uint32_t g    = (c - 256u) >> 4;             // channel group 0..15
    const uint32_t i0p  = h >> 4;                      // output h-tile index
    const uint32_t i1   = h & 15u;                     // within-tile row
    const uint32_t kin  = (g + i0p * 4u + j0p) & 15u;  // input tile (rotated)
    const uint32_t hin  = (kin >> 2) * 16u + i1;       // input row
    const uint32_t win  = (kin & 3u) * 16u;            // input segment start
    inOff = (bc * 64u + hin) * 64u + win;
  }

  const v4f* __restrict__ src = reinterpret_cast<const v4f*>(in + inOff);
  v4f* dst = reinterpret_cast<v4f*>(out + outOff);

  // 64B contiguous read ...
  v4f q0 = src[0];
  v4f q1 = src[1];
  v4f q2 = src[2];
  v4f q3 = src[3];
  // ... 64B contiguous streaming write (output is never re-read on-device).
  __builtin_nontemporal_store(q0, dst + 0);
  __builtin_nontemporal_store(q1, dst + 1);
  __builtin_nontemporal_store(q2, dst + 2);
  __builtin_nontemporal_store(q3, dst + 3);
}

extern "C" void kernel_launch(void* const* d_in, const int* in_sizes, int n_in,
                              void* d_out, int out_size, void* d_ws, size_t ws_size,
                              hipStream_t stream) {
  const float* x = (const float*)d_in[0];
  float* out = (float*)d_out;

  // 16 * 512 * 64 * 64 floats = 2,097,152 segments of 16 floats.
  constexpr uint32_t kSegments = 16u * 512u * 64u * 4u;
  constexpr uint32_t kBlock = 256u;
  global_shift2d_v2_kernel<<<dim3(kSegments / kBlock), dim3(kBlock), 0, stream>>>(x, out);
  (void)in_sizes; (void)n_in; (void)out_size; (void)d_ws; (void)ws_size;
}